// DCRNNCell_79620103733873
// MI455X (gfx1250) — compile-verified
//
#include <hip/hip_runtime.h>
#include <hip/hip_bf16.h>
#include <math.h>

// ---------------------------------------------------------------------------
// DCRNN cell for MI455X (gfx1250, wave32).
//  - Diffusion graph conv: block-per-edge f32 global atomics (dest ~L2-resident)
//  - GEMMs: V_WMMA_F32_16X16X4_F32 (full f32 accuracy, matches reference dtype)
//    with LDS-staged K-chunks (bank-conflict-free padded stride).
// Workspace layout (floats): X0 | X1 | X2 | ZR, each M*128 = 40,960,000 floats
// -> requires ~656 MB of d_ws.
// ---------------------------------------------------------------------------

#define N_NODES 10000
#define BATCH   32
#define N_EDGES 160000
#define DIMF    128              // concat feature width (DIM_IN + DIM_OUT)
#define DIMO    64
#define MROWS   (BATCH * N_NODES)          // 320000, divisible by 64
#define ROWF    ((size_t)MROWS * DIMF)     // floats per X buffer

typedef __attribute__((ext_vector_type(2))) float v2f;
typedef __attribute__((ext_vector_type(8))) float v8f;

// ---------------- concat / init / scatter ----------------------------------

__global__ __launch_bounds__(256)
void concat0_kernel(const float* __restrict__ in, const float* __restrict__ hx,
                    float* __restrict__ X0) {
  long idx = (long)blockIdx.x * blockDim.x + threadIdx.x;   // float4 index
  long t = idx * 4;
  long gm = t >> 7;                 // row
  int  j  = (int)(t & 127);         // feature (multiple of 4)
  float4 v;
  if (j < 64) v = *(const float4*)&in[gm * 64 + j];
  else        v = *(const float4*)&hx[gm * 64 + (j - 64)];
  *(float4*)&X0[t] = v;
}

// X0' = concat(inputs, r * hx), r = sigmoid-gate stored in ZR[:,64:128]
__global__ __launch_bounds__(256)
void concatc_kernel(const float* __restrict__ in, const float* __restrict__ hx,
                    const float* __restrict__ ZR, float* __restrict__ X0) {
  long idx = (long)blockIdx.x * blockDim.x + threadIdx.x;
  long t = idx * 4;
  long gm = t >> 7;
  int  j  = (int)(t & 127);
  float4 v;
  if (j < 64) {
    v = *(const float4*)&in[gm * 64 + j];
  } else {
    int f = j - 64;
    float4 h = *(const float4*)&hx[gm * 64 + f];
    float4 r = *(const float4*)&ZR[gm * 128 + 64 + f];
    v.x = h.x * r.x; v.y = h.y * r.y; v.z = h.z * r.z; v.w = h.w * r.w;
  }
  *(float4*)&X0[t] = v;
}

// X1 = 0 ; X2 = -X0   (scatter then adds w*A*X0 into X1 and 2w*A*X1 into X2)
__global__ __launch_bounds__(256)
void init_kernel(const float* __restrict__ X0, float* __restrict__ X1,
                 float* __restrict__ X2) {
  long idx = (long)blockIdx.x * blockDim.x + threadIdx.x;
  float4 z = make_float4(0.f, 0.f, 0.f, 0.f);
  float4 v = ((const float4*)X0)[idx];
  float4 nv = make_float4(-v.x, -v.y, -v.z, -v.w);
  ((float4*)X1)[idx] = z;
  ((float4*)X2)[idx] = nv;
}

// y[b, row[e], :] += scale * w[e] * x[b, col[e], :]   for all b, features
__global__ __launch_bounds__(256)
void scatter_kernel(const float* __restrict__ x, float* __restrict__ y,
                    const int* __restrict__ rows, const int* __restrict__ cols,
                    const float* __restrict__ ew, float scale) {
  const int e = blockIdx.x;
  const long r = rows[e];
  const long c = cols[e];
  const float s = scale * ew[e];
  for (int idx = threadIdx.x; idx < BATCH * DIMF; idx += 256) {
    int b = idx >> 7;              // batch
    int f = idx & 127;             // feature
    long src = ((long)b * N_NODES + c) * DIMF + f;
    long dst = ((long)b * N_NODES + r) * DIMF + f;
    atomicAdd(&y[dst], s * x[src]);
  }
}

// ---------------- WMMA f32 GEMMs -------------------------------------------
// Each wave computes a 16-row tile across all output columns.
// K = 384 processed as 12 chunks of 32 (chunk c: X[c/4], feats (c%4)*32..).
// LDS: W chunk [G][32] and A chunk [64][32], both padded to stride 34 floats
// so ds_load_b64 from 16 lanes hits distinct bank pairs.
//
// Fragment layouts (CDNA5 ISA 7.12.2, 32-bit 16x4 A / 4x16 B / 16x16 C):
//   A: lane t (m = t&15): VGPR0 = A[m][k + 2*(t>>4)], VGPR1 = +1
//   B: lane t (n = t&15): VGPR0 = B[k + 2*(t>>4)][n], VGPR1 = +1
//   C/D: VGPR v, lane t: row = v + 8*(t>>4), col = t&15

#define KC   32
#define AST  34
#define WST  34

__global__ __launch_bounds__(128)
void gemm_gates_kernel(const float* __restrict__ X0, const float* __restrict__ X1,
                       const float* __restrict__ X2, const float* __restrict__ Wg,
                       const float* __restrict__ bg, float* __restrict__ ZR) {
  __shared__ float wlds[128 * WST];   // 17408 B
  __shared__ float alds[64 * AST];    //  8704 B
  const int tid  = threadIdx.x;
  const int wv   = tid >> 5;
  const int lane = tid & 31;
  const int ln   = lane & 15;
  const int hi   = lane >> 4;
  const long mbase = (long)blockIdx.x * 64;

  v8f acc[8];
#pragma unroll
  for (int i = 0; i < 8; ++i) acc[i] = (v8f){0, 0, 0, 0, 0, 0, 0, 0};

  for (int kb = 0; kb < 12; ++kb) {
    const int kbase = kb * KC;
    const float* Xsrc = (kb < 4) ? X0 : (kb < 8) ? X1 : X2;
    const int feat0 = kbase & 127;
    for (int idx = tid; idx < 128 * KC; idx += 128) {      // stage W chunk
      int g = idx >> 5, kc = idx & 31;
      wlds[g * WST + kc] = Wg[g * 384 + kbase + kc];
    }
    for (int idx = tid; idx < 64 * KC; idx += 128) {       // stage A chunk
      int r = idx >> 5, kc = idx & 31;
      alds[r * AST + kc] = Xsrc[(mbase + r) * DIMF + feat0 + kc];
    }
    __syncthreads();
#pragma unroll
    for (int k = 0; k < KC; k += 4) {
      v2f a = *(const v2f*)&alds[(wv * 16 + ln) * AST + k + 2 * hi];
#pragma unroll
      for (int n0 = 0; n0 < 8; ++n0) {
        v2f b = *(const v2f*)&wlds[(n0 * 16 + ln) * WST + k + 2 * hi];
        acc[n0] = __builtin_amdgcn_wmma_f32_16x16x4_f32(
            false, a, false, b, (short)0, acc[n0], false, false);
      }
    }
    __syncthreads();
  }

  const long rowb = mbase + wv * 16;
#pragma unroll
  for (int n0 = 0; n0 < 8; ++n0) {
    const int col = n0 * 16 + ln;
    const float bias = bg[col];
#pragma unroll
    for (int v = 0; v < 8; ++v) {
      long gm = rowb + v + 8 * hi;
      float g = acc[n0][v] + bias;
      ZR[gm * 128 + col] = 1.0f / (1.0f + expf(-g));       // sigmoid
    }
  }
}

__global__ __launch_bounds__(128)
void gemm_cand_kernel(const float* __restrict__ X0, const float* __restrict__ X1,
                      const float* __restrict__ X2, const float* __restrict__ Wc,
                      const float* __restrict__ bc, const float* __restrict__ ZR,
                      const float* __restrict__ hx, float* __restrict__ out) {
  __shared__ float wlds[64 * WST];
  __shared__ float alds[64 * AST];
  const int tid  = threadIdx.x;
  const int wv   = tid >> 5;
  const int lane = tid & 31;
  const int ln   = lane & 15;
  const int hi   = lane >> 4;
  const long mbase = (long)blockIdx.x * 64;

  v8f acc[4];
#pragma unroll
  for (int i = 0; i < 4; ++i) acc[i] = (v8f){0, 0, 0, 0, 0, 0, 0, 0};

  for (int kb = 0; kb < 12; ++kb) {
    const int kbase = kb * KC;
    const float* Xsrc = (kb < 4) ? X0 : (kb < 8) ? X1 : X2;
    const int feat0 = kbase & 127;
    for (int idx = tid; idx < 64 * KC; idx += 128) {       // stage W chunk
      int g = idx >> 5, kc = idx & 31;
      wlds[g * WST + kc] = Wc[g * 384 + kbase + kc];
    }
    for (int idx = tid; idx < 64 * KC; idx += 128) {       // stage A chunk
      int r = idx >> 5, kc = idx & 31;
      alds[r * AST + kc] = Xsrc[(mbase + r) * DIMF + feat0 + kc];
    }
    __syncthreads();
#pragma unroll
    for (int k = 0; k < KC; k += 4) {
      v2f a = *(const v2f*)&alds[(wv * 16 + ln) * AST + k + 2 * hi];
#pragma unroll
      for (int n0 = 0; n0 < 4; ++n0) {
        v2f b = *(const v2f*)&wlds[(n0 * 16 + ln) * WST + k + 2 * hi];
        acc[n0] = __builtin_amdgcn_wmma_f32_16x16x4_f32(
            false, a, false, b, (short)0, acc[n0], false, false);
      }
    }
    __syncthreads();
  }

  const long rowb = mbase + wv * 16;
#pragma unroll
  for (int n0 = 0; n0 < 4; ++n0) {
    const int col = n0 * 16 + ln;                // 0..63
    const float bias = bc[col];
#pragma unroll
    for (int v = 0; v < 8; ++v) {
      long gm = rowb + v + 8 * hi;
      float cand = tanhf(acc[n0][v] + bias);
      float z = ZR[gm * 128 + col];              // z gate (first 64 cols)
      float h = hx[gm * 64 + col];
      out[gm * 64 + col] = (1.0f - z) * h + z * cand;
    }
  }
}

// ---------------- launch ----------------------------------------------------

extern "C" void kernel_launch(void* const* d_in, const int* in_sizes, int n_in,
                              void* d_out, int out_size, void* d_ws, size_t ws_size,
                              hipStream_t stream) {
  const float* inputs = (const float*)d_in[0];
  const float* hx     = (const float*)d_in[1];
  const int*   eidx   = (const int*)d_in[2];   // (2, N_EDGES) int32
  const float* ew     = (const float*)d_in[3];
  const float* Wg     = (const float*)d_in[4]; // (128, 384)
  const float* Wc     = (const float*)d_in[5]; // (64, 384)
  const float* bg     = (const float*)d_in[6];
  const float* bc     = (const float*)d_in[7];
  float* out = (float*)d_out;

  const int* erow = eidx;
  const int* ecol = eidx + N_EDGES;

  float* X0 = (float*)d_ws;
  float* X1 = X0 + ROWF;
  float* X2 = X1 + ROWF;
  float* ZR = X2 + ROWF;

  const int nVec = (int)(ROWF / 4 / 256);   // 40000 blocks of 256 float4-threads
  const int nGemm = MROWS / 64;             // 5000 blocks

  // --- phase 1: gates ---
  concat0_kernel<<<nVec, 256, 0, stream>>>(inputs, hx, X0);
  init_kernel<<<nVec, 256, 0, stream>>>(X0, X1, X2);
  scatter_kernel<<<N_EDGES, 256, 0, stream>>>(X0, X1, erow, ecol, ew, 1.0f);
  scatter_kernel<<<N_EDGES, 256, 0, stream>>>(X1, X2, erow, ecol, ew, 2.0f);
  gemm_gates_kernel<<<nGemm, 128, 0, stream>>>(X0, X1, X2, Wg, bg, ZR);

  // --- phase 2: candidate + blend ---
  concatc_kernel<<<nVec, 256, 0, stream>>>(inputs, hx, ZR, X0);
  init_kernel<<<nVec, 256, 0, stream>>>(X0, X1, X2);
  scatter_kernel<<<N_EDGES, 256, 0, stream>>>(X0, X1, erow, ecol, ew, 1.0f);
  scatter_kernel<<<N_EDGES, 256, 0, stream>>>(X1, X2, erow, ecol, ew, 2.0f);
  gemm_cand_kernel<<<nGemm, 128, 0, stream>>>(X0, X1, X2, Wc, bc, ZR, hx, out);
}